// AttnHGCN_38706245272387
// MI455X (gfx1250) — compile-verified
//
#include <hip/hip_runtime.h>
#include <math.h>

// ---------- types ----------
typedef float v2f __attribute__((ext_vector_type(2)));
typedef float v8f __attribute__((ext_vector_type(8)));

#define EMB 64
#define NHEADS 2
#define DK 32
#define INV_SQRT_DK 0.17677669529663687f  // 1/sqrt(32)

// monotone float <-> uint encoding for atomicMax-based segment max
__device__ __forceinline__ unsigned fenc(float f) {
  unsigned u = __float_as_uint(f);
  return (u & 0x80000000u) ? ~u : (u | 0x80000000u);
}
__device__ __forceinline__ float fdec(unsigned u) {
  return (u & 0x80000000u) ? __uint_as_float(u & 0x7FFFFFFFu)
                           : __uint_as_float(~u);
}

// ---------- init: copy embeddings into residual accumulators + e_cur ----------
__global__ void init_copy_kernel(const float* __restrict__ user_emb,
                                 const float* __restrict__ entity_emb,
                                 float* __restrict__ e_res, float* __restrict__ u_res,
                                 float* __restrict__ e_cur, int ne_elems, int nu_elems) {
  int stride = gridDim.x * blockDim.x;
  for (int j = blockIdx.x * blockDim.x + threadIdx.x; j < ne_elems; j += stride) {
    float v = entity_emb[j];
    e_res[j] = v;
    e_cur[j] = v;
  }
  for (int j = blockIdx.x * blockDim.x + threadIdx.x; j < nu_elems; j += stride) {
    u_res[j] = user_emb[j];
  }
}

// ---------- WMMA fp32 GEMM: P[nrows x 64] = E[nrows x 64] @ WQ[64 x 64] ----------
// One wave computes a 16-row slab across all 4 column tiles, chaining
// 16 x V_WMMA_F32_16X16X4_F32 (K=4 each) per tile. A fragment reused over N.
__global__ void __launch_bounds__(256) gemm_wmma_f32_kernel(
    const float* __restrict__ E, const float* __restrict__ WQ,
    float* __restrict__ P, int ntiles /* nrows/16 */) {
  int wid  = (blockIdx.x * blockDim.x + threadIdx.x) >> 5;
  int lane = threadIdx.x & 31;
  if (wid >= ntiles) return;                 // uniform per wave -> EXEC all-1s inside
  const int mbase = wid * 16;
  const int half  = lane >> 4;               // 0: K even pair, 1: K odd pair
  const int lrow  = lane & 15;

  v8f acc0 = {0.f,0.f,0.f,0.f,0.f,0.f,0.f,0.f};
  v8f acc1 = acc0, acc2 = acc0, acc3 = acc0;

  const float* arow = E + (size_t)(mbase + lrow) * EMB;

#pragma unroll
  for (int ks = 0; ks < 16; ++ks) {
    const int k0 = ks * 4 + half * 2;
    // A fragment (16x4 f32): lanes 0-15 rows M=0..15 K={k0,k0+1}; lanes 16-31 K={k0+2,k0+3}
    v2f a;
    a.x = arow[k0];
    a.y = arow[k0 + 1];
    // B fragment (4x16 f32): lanes 0-15 N=0..15 K=k0 / lanes 16-31 K=k0+2 (mirrors A/C split)
    v2f b0, b1, b2, b3;
    b0.x = WQ[k0 * EMB + ( 0 + lrow)];  b0.y = WQ[(k0 + 1) * EMB + ( 0 + lrow)];
    b1.x = WQ[k0 * EMB + (16 + lrow)];  b1.y = WQ[(k0 + 1) * EMB + (16 + lrow)];
    b2.x = WQ[k0 * EMB + (32 + lrow)];  b2.y = WQ[(k0 + 1) * EMB + (32 + lrow)];
    b3.x = WQ[k0 * EMB + (48 + lrow)];  b3.y = WQ[(k0 + 1) * EMB + (48 + lrow)];

    acc0 = __builtin_amdgcn_wmma_f32_16x16x4_f32(false, a, false, b0, (short)0, acc0, false, false);
    acc1 = __builtin_amdgcn_wmma_f32_16x16x4_f32(false, a, false, b1, (short)0, acc1, false, false);
    acc2 = __builtin_amdgcn_wmma_f32_16x16x4_f32(false, a, false, b2, (short)0, acc2, false, false);
    acc3 = __builtin_amdgcn_wmma_f32_16x16x4_f32(false, a, false, b3, (short)0, acc3, false, false);
  }

  // C/D layout: VGPR r, lanes 0-15 -> M = r ; lanes 16-31 -> M = 8 + r ; N = ntile*16 + lrow
  const int rowoff = mbase + half * 8;
#pragma unroll
  for (int r = 0; r < 8; ++r) {
    float* prow = P + (size_t)(rowoff + r) * EMB + lrow;
    prow[0]  = acc0[r];
    prow[16] = acc1[r];
    prow[32] = acc2[r];
    prow[48] = acc3[r];
  }
}

// ---------- pass 1: per-edge attention scores + segment max (8 lanes/edge) ----------
__global__ void score_kernel(const float* __restrict__ proj,
                             const float* __restrict__ rel_emb,
                             const int* __restrict__ head, const int* __restrict__ tail,
                             const int* __restrict__ etype,
                             unsigned* __restrict__ smax, float* __restrict__ score,
                             int nedges) {
  int t = blockIdx.x * blockDim.x + threadIdx.x;
  int e = t >> 3;
  if (e >= nedges) return;
  int sl = t & 7;                       // lanes 0-3 -> head 0, lanes 4-7 -> head 1
  int h  = head[e];
  int ta = tail[e];
  int r  = etype[e] - 1;

  const float4* q4 = (const float4*)(proj    + (size_t)h  * EMB + sl * 8);
  const float4* k4 = (const float4*)(proj    + (size_t)ta * EMB + sl * 8);
  const float4* r4 = (const float4*)(rel_emb + (size_t)r  * EMB + sl * 8);
  float4 qa = q4[0], qb = q4[1];
  float4 ka = k4[0], kb = k4[1];
  float4 ra = r4[0], rb = r4[1];

  float dot = qa.x * ka.x * ra.x + qa.y * ka.y * ra.y + qa.z * ka.z * ra.z + qa.w * ka.w * ra.w
            + qb.x * kb.x * rb.x + qb.y * kb.y * rb.y + qb.z * kb.z * rb.z + qb.w * kb.w * rb.w;
  // reduce across the 4 lanes covering one head (groups of 4 are lane-aligned)
  dot += __shfl_xor(dot, 1, 32);
  dot += __shfl_xor(dot, 2, 32);

  if ((sl & 3) == 0) {
    int hh = sl >> 2;
    float s = dot * INV_SQRT_DK;
    score[(size_t)e * NHEADS + hh] = s;
    atomicMax(&smax[(size_t)h * NHEADS + hh], fenc(s));
  }
}

// ---------- pass 2: exp(score - segmax) + segment sum; overwrite score with ex ----------
__global__ void exp_kernel(const int* __restrict__ head,
                           const unsigned* __restrict__ smax,
                           float* __restrict__ ssum, float* __restrict__ score,
                           int nedges) {
  int t = blockIdx.x * blockDim.x + threadIdx.x;
  if (t >= nedges * NHEADS) return;
  int e  = t >> 1;
  int hh = t & 1;
  int h  = head[e];
  float m  = fdec(smax[(size_t)h * NHEADS + hh]);
  float ex = __expf(score[t] - m);
  score[t] = ex;
  atomicAdd(&ssum[(size_t)h * NHEADS + hh], ex);
}

// ---------- pass 3: attn-weighted message scatter into e_agg (8 lanes/edge) ----------
__global__ void agg_kernel(const float* __restrict__ e_cur,
                           const float* __restrict__ rel_emb,
                           const int* __restrict__ head, const int* __restrict__ tail,
                           const int* __restrict__ etype,
                           const float* __restrict__ ex, const float* __restrict__ ssum,
                           float* __restrict__ e_agg, int nedges) {
  int t = blockIdx.x * blockDim.x + threadIdx.x;
  int e = t >> 3;
  if (e >= nedges) return;
  int sl = t & 7;
  int h  = head[e];
  int ta = tail[e];
  int r  = etype[e] - 1;
  int hh = sl >> 2;
  float attn = ex[(size_t)e * NHEADS + hh] / ssum[(size_t)h * NHEADS + hh];

  const float4* v4 = (const float4*)(e_cur   + (size_t)ta * EMB + sl * 8);
  const float4* r4 = (const float4*)(rel_emb + (size_t)r  * EMB + sl * 8);
  float4 va = v4[0], vb = v4[1];
  float4 ra = r4[0], rb = r4[1];

  float* dst = e_agg + (size_t)h * EMB + sl * 8;
  atomicAdd(dst + 0, va.x * ra.x * attn);
  atomicAdd(dst + 1, va.y * ra.y * attn);
  atomicAdd(dst + 2, va.z * ra.z * attn);
  atomicAdd(dst + 3, va.w * ra.w * attn);
  atomicAdd(dst + 4, vb.x * rb.x * attn);
  atomicAdd(dst + 5, vb.y * rb.y * attn);
  atomicAdd(dst + 6, vb.z * rb.z * attn);
  atomicAdd(dst + 7, vb.w * rb.w * attn);
}

// ---------- user aggregation from interaction edges (8 lanes/edge) ----------
__global__ void user_agg_kernel(const float* __restrict__ e_cur,
                                const int* __restrict__ iu, const int* __restrict__ ii,
                                const float* __restrict__ w,
                                float* __restrict__ u_agg, int nedges) {
  int t = blockIdx.x * blockDim.x + threadIdx.x;
  int e = t >> 3;
  if (e >= nedges) return;
  int sl = t & 7;
  int u  = iu[e];
  int it = ii[e];
  float ww = w[e];

  const float4* v4 = (const float4*)(e_cur + (size_t)it * EMB + sl * 8);
  float4 va = v4[0], vb = v4[1];

  float* dst = u_agg + (size_t)u * EMB + sl * 8;
  atomicAdd(dst + 0, ww * va.x);
  atomicAdd(dst + 1, ww * va.y);
  atomicAdd(dst + 2, ww * va.z);
  atomicAdd(dst + 3, ww * va.w);
  atomicAdd(dst + 4, ww * vb.x);
  atomicAdd(dst + 5, ww * vb.y);
  atomicAdd(dst + 6, ww * vb.z);
  atomicAdd(dst + 7, ww * vb.w);
}

// ---------- fused l2norm + residual add (one wave per 64-float row) ----------
__global__ void norm_res_kernel(const float* __restrict__ agg,
                                float* __restrict__ res, float* __restrict__ cur,
                                int nrows) {
  int wid  = (blockIdx.x * blockDim.x + threadIdx.x) >> 5;
  int lane = threadIdx.x & 31;
  if (wid >= nrows) return;
  const float2* a = (const float2*)(agg + (size_t)wid * EMB);
  float2 v = a[lane];
  float ss = v.x * v.x + v.y * v.y;
#pragma unroll
  for (int o = 16; o > 0; o >>= 1) ss += __shfl_xor(ss, o, 32);
  float inv = 1.0f / fmaxf(sqrtf(ss), 1e-12f);
  float nx = v.x * inv, ny = v.y * inv;
  float2* rp = (float2*)(res + (size_t)wid * EMB);
  float2 old = rp[lane];
  rp[lane] = make_float2(old.x + nx, old.y + ny);
  if (cur) {
    ((float2*)(cur + (size_t)wid * EMB))[lane] = make_float2(nx, ny);
  }
}

// ---------- host-side orchestration ----------
extern "C" void kernel_launch(void* const* d_in, const int* in_sizes, int n_in,
                              void* d_out, int out_size, void* d_ws, size_t ws_size,
                              hipStream_t stream) {
  const float* user_emb   = (const float*)d_in[0];
  const float* entity_emb = (const float*)d_in[1];
  const int*   edge_index = (const int*)d_in[2];   // [2, Ekg]
  const int*   edge_type  = (const int*)d_in[3];   // [Ekg]
  const int*   inter_edge = (const int*)d_in[4];   // [2, Eui]
  const float* inter_w    = (const float*)d_in[5]; // [Eui]
  const float* W_Q        = (const float*)d_in[6]; // [64, 64]
  const float* rel_emb    = (const float*)d_in[7]; // [16, 64]

  const int n_usr = in_sizes[0] / EMB;
  const int n_ent = in_sizes[1] / EMB;
  const int nkg   = in_sizes[3];
  const int nui   = in_sizes[5];

  const int* kg_head = edge_index;
  const int* kg_tail = edge_index + nkg;
  const int* ui_user = inter_edge;
  const int* ui_item = inter_edge + nui;

  // workspace carve-out (floats)
  float* ws    = (float*)d_ws;
  float* proj  = ws;                                   // n_ent*64
  float* e_cur = proj  + (size_t)n_ent * EMB;          // n_ent*64
  float* e_agg = e_cur + (size_t)n_ent * EMB;          // n_ent*64   -- zeroed
  float* u_agg = e_agg + (size_t)n_ent * EMB;          // n_usr*64   -- zeroed
  float* ssum  = u_agg + (size_t)n_usr * EMB;          // n_ent*2    -- zeroed
  unsigned* smax = (unsigned*)(ssum + (size_t)n_ent * NHEADS); // n_ent*2 -- zeroed
  float* score = (float*)(smax + (size_t)n_ent * NHEADS);      // nkg*2

  float* e_res = (float*)d_out;                        // n_ent*64
  float* u_res = e_res + (size_t)n_ent * EMB;          // n_usr*64

  const int B = 256;
  const size_t zero_bytes =
      ((size_t)n_ent * EMB + (size_t)n_usr * EMB + (size_t)n_ent * NHEADS * 2) * sizeof(float);

  // residual init + e_cur = entity_emb
  init_copy_kernel<<<2048, B, 0, stream>>>(user_emb, entity_emb, e_res, u_res, e_cur,
                                           n_ent * EMB, n_usr * EMB);

  const int ntiles      = n_ent / 16;                  // 100000/16 = 6250 (exact)
  const int gemm_blocks = (ntiles * 32 + B - 1) / B;
  const int sc_blocks   = (int)(((size_t)nkg * 8 + B - 1) / B);
  const int ex_blocks   = (int)(((size_t)nkg * NHEADS + B - 1) / B);
  const int ui_blocks   = (int)(((size_t)nui * 8 + B - 1) / B);
  const int ne_wblocks  = (n_ent * 32 + B - 1) / B;
  const int nu_wblocks  = (n_usr * 32 + B - 1) / B;

  for (int hop = 0; hop < 2; ++hop) {
    // zero accumulators/stats (contiguous region: e_agg, u_agg, ssum, smax)
    hipMemsetAsync(e_agg, 0, zero_bytes, stream);

    // proj = e_cur @ W_Q  (V_WMMA_F32_16X16X4_F32)
    gemm_wmma_f32_kernel<<<gemm_blocks, B, 0, stream>>>(e_cur, W_Q, proj, ntiles);

    // attention softmax over head-segments
    score_kernel<<<sc_blocks, B, 0, stream>>>(proj, rel_emb, kg_head, kg_tail, edge_type,
                                              smax, score, nkg);
    exp_kernel<<<ex_blocks, B, 0, stream>>>(kg_head, smax, ssum, score, nkg);
    agg_kernel<<<sc_blocks, B, 0, stream>>>(e_cur, rel_emb, kg_head, kg_tail, edge_type,
                                            score, ssum, e_agg, nkg);

    // user aggregation from interaction edges
    user_agg_kernel<<<ui_blocks, B, 0, stream>>>(e_cur, ui_user, ui_item, inter_w, u_agg, nui);

    // normalize + residual; e_cur updated for next hop
    norm_res_kernel<<<ne_wblocks, B, 0, stream>>>(e_agg, e_res, e_cur, n_ent);
    norm_res_kernel<<<nu_wblocks, B, 0, stream>>>(u_agg, u_res, (float*)nullptr, n_usr);
  }
}